// RelationalGraphLayer_33998961115151
// MI455X (gfx1250) — compile-verified
//
#include <hip/hip_runtime.h>

#define N_NODES 50000
#define N_EDGES 800000
#define DIM 128
#define NTYPES 3
#define AGG_DIM (DIM + NTYPES * DIM) /* 512 */

typedef __attribute__((ext_vector_type(2))) float v2f;
typedef __attribute__((ext_vector_type(8))) float v8f;

// -------- Kernel 1: build agg[:,0:128] = relu(x), zero mailbox cols --------
__global__ void rgl_init_agg(const float* __restrict__ nf, float* __restrict__ agg) {
    size_t i = (size_t)blockIdx.x * blockDim.x + threadIdx.x;
    const size_t total = (size_t)N_NODES * AGG_DIM;
    if (i >= total) return;
    const int c = (int)(i & (AGG_DIM - 1));
    const size_t n = i >> 9;  // /512
    float v = 0.0f;
    if (c < DIM) {
        const float x = nf[n * DIM + c];
        v = x > 0.0f ? x : 0.0f;
    }
    agg[i] = v;
}

// -------- Kernel 2: per-edge-type message GEMM + ReLU + masked scatter-add --------
// One wave per 16-edge tile. 800000/16 = 50000 tiles = gridDim(6250) * 8 waves.
__global__ __launch_bounds__(256) void rgl_edge_msg(
    const float* __restrict__ nf, const float* __restrict__ We,
    const float* __restrict__ be, const int* __restrict__ esrc,
    const int* __restrict__ edst, const int* __restrict__ etyp,
    float* __restrict__ agg) {
    __shared__ __align__(16) float lds[8][16 * DIM];  // 64 KB/block

    const int lane  = threadIdx.x & 31;
    const int wave  = threadIdx.x >> 5;
    const int tile  = blockIdx.x * 8 + wave;  // always valid (exact grid)
    const int e0    = tile * 16;
    const int r16   = lane & 15;
    const int khalf = lane >> 4;  // 0 or 1

    const int mySrc = esrc[e0 + r16];
    const int myDst = edst[e0 + r16];
    const int myTyp = etyp[e0 + r16];

    float* slab = &lds[wave][0];
    // Stage gathered 16x128 A tile into LDS: row it, 32 lanes x float4.
    #pragma unroll 4
    for (int it = 0; it < 16; ++it) {
        const int sRow = __shfl(mySrc, it, 32);
        const float4 v = *reinterpret_cast<const float4*>(
            nf + (size_t)sRow * DIM + lane * 4);
        *reinterpret_cast<float4*>(slab + it * DIM + lane * 4) = v;
    }
    __syncthreads();

    for (int t = 0; t < NTYPES; ++t) {
        const float* Wt = We + (size_t)t * DIM * DIM;  // W_e[t][n][k]
        for (int ct = 0; ct < 8; ++ct) {
            const int n = ct * 16 + r16;  // output column for this lane
            const float bv = be[t * DIM + n];
            v8f acc;
            #pragma unroll
            for (int j = 0; j < 8; ++j) acc[j] = bv;

            const float* brow = Wt + (size_t)n * DIM;
            const float* aro  = slab + r16 * DIM;
            #pragma unroll 4
            for (int kk = 0; kk < 32; ++kk) {
                const int kb = kk * 4 + khalf * 2;
                const v2f a = *reinterpret_cast<const v2f*>(aro + kb);
                const v2f b = *reinterpret_cast<const v2f*>(brow + kb);
                acc = __builtin_amdgcn_wmma_f32_16x16x4_f32(
                    false, a, false, b, (short)0, acc, false, false);
            }
            // ReLU + scatter-add only for matching edge type.
            #pragma unroll
            for (int j = 0; j < 8; ++j) {
                const int m   = j + khalf * 8;          // edge row in tile
                const int etm = __shfl(myTyp, m, 32);   // full EXEC here
                const int dm  = __shfl(myDst, m, 32);   // full EXEC here
                float v = acc[j];
                v = v > 0.0f ? v : 0.0f;
                if (etm == t) {
                    atomicAdd(agg + (size_t)dm * AGG_DIM + DIM + t * DIM + n, v);
                }
            }
        }
    }
}

// -------- Kernel 3: per-node-type updater GEMM + select + residual --------
// One wave per 16-node tile (3125 tiles).
__global__ __launch_bounds__(256) void rgl_node_update(
    const float* __restrict__ nf, const float* __restrict__ Wn,
    const float* __restrict__ bn, const int* __restrict__ ntyp,
    const float* __restrict__ agg, float* __restrict__ out) {
    const int lane = threadIdx.x & 31;
    const int wave = threadIdx.x >> 5;
    const int tile = blockIdx.x * 8 + wave;
    if (tile >= (N_NODES / 16)) return;  // wave-uniform exit; no LDS used
    const int n0    = tile * 16;
    const int r16   = lane & 15;
    const int khalf = lane >> 4;

    const int myNt = ntyp[n0 + r16];
    const float* arow = agg + (size_t)(n0 + r16) * AGG_DIM;

    for (int ct = 0; ct < 8; ++ct) {
        const int c = ct * 16 + r16;  // output column for this lane
        v8f acc0, acc1, acc2;
        {
            const float b0 = bn[0 * DIM + c];
            const float b1 = bn[1 * DIM + c];
            const float b2 = bn[2 * DIM + c];
            #pragma unroll
            for (int j = 0; j < 8; ++j) { acc0[j] = b0; acc1[j] = b1; acc2[j] = b2; }
        }
        const float* w0 = Wn + (size_t)(0 * DIM + c) * AGG_DIM;
        const float* w1 = Wn + (size_t)(1 * DIM + c) * AGG_DIM;
        const float* w2 = Wn + (size_t)(2 * DIM + c) * AGG_DIM;
        #pragma unroll 4
        for (int kk = 0; kk < AGG_DIM / 4; ++kk) {
            const int kb = kk * 4 + khalf * 2;
            const v2f a   = *reinterpret_cast<const v2f*>(arow + kb);
            const v2f bb0 = *reinterpret_cast<const v2f*>(w0 + kb);
            const v2f bb1 = *reinterpret_cast<const v2f*>(w1 + kb);
            const v2f bb2 = *reinterpret_cast<const v2f*>(w2 + kb);
            acc0 = __builtin_amdgcn_wmma_f32_16x16x4_f32(false, a, false, bb0, (short)0, acc0, false, false);
            acc1 = __builtin_amdgcn_wmma_f32_16x16x4_f32(false, a, false, bb1, (short)0, acc1, false, false);
            acc2 = __builtin_amdgcn_wmma_f32_16x16x4_f32(false, a, false, bb2, (short)0, acc2, false, false);
        }
        #pragma unroll
        for (int j = 0; j < 8; ++j) {
            const int m    = j + khalf * 8;        // node row in tile
            const int ntm  = __shfl(myNt, m, 32);  // full EXEC
            const int node = n0 + m;
            float v = (ntm == 0) ? acc0[j] : (ntm == 1) ? acc1[j] : acc2[j];
            v += nf[(size_t)node * DIM + ct * 16 + r16];
            out[(size_t)node * DIM + ct * 16 + r16] = v;
        }
    }
}

extern "C" void kernel_launch(void* const* d_in, const int* in_sizes, int n_in,
                              void* d_out, int out_size, void* d_ws, size_t ws_size,
                              hipStream_t stream) {
    const float* nf   = (const float*)d_in[0];  // node_feature [N,128]
    const float* We   = (const float*)d_in[1];  // W_e [3,128,128]
    const float* be   = (const float*)d_in[2];  // b_e [3,128]
    const float* Wn   = (const float*)d_in[3];  // W_n [3,128,512]
    const float* bn   = (const float*)d_in[4];  // b_n [3,128]
    const int*   esrc = (const int*)d_in[5];
    const int*   edst = (const int*)d_in[6];
    const int*   etyp = (const int*)d_in[7];
    const int*   ntyp = (const int*)d_in[8];
    float* out = (float*)d_out;
    float* agg = (float*)d_ws;  // [N, 512] f32 = 102.4 MB (L2-resident)

    {
        const size_t total = (size_t)N_NODES * AGG_DIM;
        const int blocks = (int)((total + 255) / 256);
        hipLaunchKernelGGL(rgl_init_agg, dim3(blocks), dim3(256), 0, stream, nf, agg);
    }
    hipLaunchKernelGGL(rgl_edge_msg, dim3(N_EDGES / 16 / 8), dim3(256), 0, stream,
                       nf, We, be, esrc, edst, etyp, agg);
    hipLaunchKernelGGL(rgl_node_update, dim3((N_NODES / 16 + 7) / 8), dim3(256), 0,
                       stream, nf, Wn, bn, ntyp, agg, out);
}